// VectorQuantizedBottleneck_63625645522984
// MI455X (gfx1250) — compile-verified
//
#include <hip/hip_runtime.h>

// CDNA5 / gfx1250, wave32.
// Scalar VQ: for each element x of `encoded`, find nearest scalar code among 512,
// via WMMA-computed score matrix  score[i][j] = c_j^2 - 2*c_j*x_i  (rank-2 matmul
// on V_WMMA_F32_16X16X4_F32). Codebook staged in LDS via async global->LDS DMA
// (GLOBAL_LOAD_ASYNC_TO_LDS_B128, tracked by ASYNCcnt).

typedef __attribute__((ext_vector_type(2))) float v2f;
typedef __attribute__((ext_vector_type(8))) float v8f;
typedef __attribute__((ext_vector_type(4))) int   v4i;

// Pointer-to-AS-qualified-int4 types for the async LDS-DMA builtin
// (declared as (int4 addrspace(1)*, int4 addrspace(3)*, imm, imm)).
typedef __attribute__((address_space(1))) v4i* gv4i_ptr;
typedef __attribute__((address_space(3))) v4i* lv4i_ptr;

#define VQ_N     (1024 * 256)
#define VQ_K     512
#define VQ_TILES (VQ_K / 16)   // 32 code tiles of 16

#if defined(__has_builtin)
#if __has_builtin(__builtin_amdgcn_global_load_async_to_lds_b128) && \
    __has_builtin(__builtin_amdgcn_s_wait_asynccnt)
#define VQ_HAVE_ASYNC_LDS 1
#endif
#endif
#ifndef VQ_HAVE_ASYNC_LDS
#define VQ_HAVE_ASYNC_LDS 0
#endif

__global__ __launch_bounds__(256)
void vq_wmma_quantize(const float* __restrict__ x,
                      const float* __restrict__ codes,
                      float* __restrict__ latent)
{
    __shared__ float lds_c[VQ_K];

#if VQ_HAVE_ASYNC_LDS
    // Async DMA fill: 512 floats = 2KB. 128 lanes x 16B via
    // GLOBAL_LOAD_ASYNC_TO_LDS_B128 (ASYNCcnt), no VGPR round-trip.
    if (threadIdx.x < (VQ_K / 4)) {
        __builtin_amdgcn_global_load_async_to_lds_b128(
            (gv4i_ptr)(codes + threadIdx.x * 4),
            (lv4i_ptr)(&lds_c[threadIdx.x * 4]),
            0, 0);
    }
    __builtin_amdgcn_s_wait_asynccnt(0);
    __syncthreads();
#else
    for (int i = threadIdx.x; i < VQ_K; i += 256) lds_c[i] = codes[i];
    __syncthreads();
#endif

    const int lane        = threadIdx.x & 31;
    const int waveInBlock = threadIdx.x >> 5;
    const int globalWave  = blockIdx.x * 8 + waveInBlock;
    const int numWaves    = gridDim.x * 8;
    const bool lower      = lane < 16;
    const int  col        = lane & 15;

    const int numGroups = VQ_N / 16;

    for (int group = globalWave; group < numGroups; group += numWaves) {
        const int base = group * 16;

        // A matrix 16x4 (only k=0,1 used): lanes 0-15 hold k=0,1; lanes 16-31 hold k=2,3 (zero).
        // A[M][0] = x_M, A[M][1] = 1.0
        float xv = lower ? x[base + lane] : 0.0f;
        v2f A;
        A.x = lower ? xv   : 0.0f;
        A.y = lower ? 1.0f : 0.0f;

        float best[8], bcode[8];
#pragma unroll
        for (int r = 0; r < 8; ++r) { best[r] = 3.0e38f; bcode[r] = 0.0f; }

#pragma unroll 4
        for (int t = 0; t < VQ_TILES; ++t) {
            // Candidate code value for this lane's column (needed in all 32 lanes).
            float c = lds_c[t * 16 + col];
            // B matrix 4x16: B[0][N] = -2*c_N, B[1][N] = c_N^2. Lanes 16-31 hold
            // k=2,3 whose A entries are zero, so their B values are don't-care.
            v2f B;
            B.x = -2.0f * c;
            B.y = c * c;

            v8f acc = {};
            // D = A x B + 0 : scores for 16 elements x 16 codes, fp32.
            acc = __builtin_amdgcn_wmma_f32_16x16x4_f32(
                false, A, false, B, (short)0, acc, false, false);

            // Running min per (row, column-lane): carry the winning code VALUE.
            // Strict '<' keeps the earliest tile on ties (argmin-first semantics).
#pragma unroll
            for (int r = 0; r < 8; ++r) {
                if (acc[r] < best[r]) { best[r] = acc[r]; bcode[r] = c; }
            }
        }

        // Butterfly min across the 16 column-lanes of each half-wave.
        // Strict '<' => lane 0 / lane 16 converge to the lowest-index winner on ties.
#pragma unroll
        for (int m = 1; m < 16; m <<= 1) {
#pragma unroll
            for (int r = 0; r < 8; ++r) {
                float os = __shfl_xor(best[r],  m, 32);
                float oc = __shfl_xor(bcode[r], m, 32);
                if (os < best[r]) { best[r] = os; bcode[r] = oc; }
            }
        }

        // Lane 0 holds rows 0..7, lane 16 holds rows 8..15. Pack into float4 and
        // emit global_store_b128 (base is a multiple of 16 floats -> 64B aligned).
        if (lane == 0) {
            float4 lo = make_float4(bcode[0], bcode[1], bcode[2], bcode[3]);
            float4 hi = make_float4(bcode[4], bcode[5], bcode[6], bcode[7]);
            float4* dst = reinterpret_cast<float4*>(latent + base);
            dst[0] = lo;
            dst[1] = hi;
        } else if (lane == 16) {
            float4 lo = make_float4(bcode[0], bcode[1], bcode[2], bcode[3]);
            float4 hi = make_float4(bcode[4], bcode[5], bcode[6], bcode[7]);
            float4* dst = reinterpret_cast<float4*>(latent + base + 8);
            dst[0] = lo;
            dst[1] = hi;
        }
    }
}

// Per-block partial sums of (latent - x)^2, fixed order -> deterministic.
__global__ __launch_bounds__(256)
void vq_loss_partial(const float* __restrict__ x,
                     const float* __restrict__ latent,
                     float* __restrict__ partial, int n)
{
    __shared__ float red[256];
    float s = 0.0f;
    for (int i = blockIdx.x * 256 + threadIdx.x; i < n; i += gridDim.x * 256) {
        float d = latent[i] - x[i];
        s += d * d;
    }
    red[threadIdx.x] = s;
    __syncthreads();
#pragma unroll
    for (int off = 128; off > 0; off >>= 1) {
        if (threadIdx.x < off) red[threadIdx.x] += red[threadIdx.x + off];
        __syncthreads();
    }
    if (threadIdx.x == 0) partial[blockIdx.x] = red[0];
}

__global__ __launch_bounds__(256)
void vq_loss_final(const float* __restrict__ partial, int nb,
                   float* __restrict__ out_loss)
{
    __shared__ float red[256];
    float s = 0.0f;
    for (int i = threadIdx.x; i < nb; i += 256) s += partial[i];
    red[threadIdx.x] = s;
    __syncthreads();
#pragma unroll
    for (int off = 128; off > 0; off >>= 1) {
        if (threadIdx.x < off) red[threadIdx.x] += red[threadIdx.x + off];
        __syncthreads();
    }
    // loss = ((latent-x)^2 + BETA*(x-latent)^2).sum(axis=1).mean(axis=0), BETA=1
    //      = 2 * total / batch(1024)
    if (threadIdx.x == 0) *out_loss = red[0] * (2.0f / 1024.0f);
}

extern "C" void kernel_launch(void* const* d_in, const int* in_sizes, int n_in,
                              void* d_out, int out_size, void* d_ws, size_t ws_size,
                              hipStream_t stream)
{
    const float* x     = (const float*)d_in[0];  // encoded  [1024,256]
    const float* codes = (const float*)d_in[1];  // embeddings [1,512]
    float* latent = (float*)d_out;               // [262144]
    float* loss   = latent + VQ_N;               // [1]
    float* partial = (float*)d_ws;               // 256 floats scratch

    (void)in_sizes; (void)n_in; (void)out_size; (void)ws_size;

    // 128 blocks x 8 waves = 1024 waves; 16384 groups -> 16 groups per wave.
    vq_wmma_quantize<<<128, 256, 0, stream>>>(x, codes, latent);
    vq_loss_partial<<<256, 256, 0, stream>>>(x, latent, partial, VQ_N);
    vq_loss_final<<<1, 256, 0, stream>>>(partial, 256, loss);
}